// SetEncoder_31336081391998
// MI455X (gfx1250) — compile-verified
//
#include <hip/hip_runtime.h>

// ============================================================================
// Set Transformer encoder for MI455X (gfx1250), fp32 WMMA (V_WMMA_F32_16X16X4_F32)
// with ASYNCcnt-tracked global->LDS double-buffered GEMM staging.
//
// Pipeline: relu(linear 16->256) -> ISAB(256) -> ISAB(512) -> ISAB(512) -> PMA
// B=32, N=2048, DH=512, H=8 heads x 64, NI=64 induced points, NS=1 seed.
//
// Input flattening assumption (setup_inputs dict insertion order):
//   0:x 1:Wl 2:bl
//   3:isab1.I   4..15:isab1.mab0   16..27:isab1.mab1
//   28:isab_a.I 29..40:isab_a.mab0 41..52:isab_a.mab1
//   53:isab_b.I 54..65:isab_b.mab0 66..77:isab_b.mab1
//   78:pma.S    79..90:pma.mab
//   (each mab: Wq,bq,Wk,bk,Wv,bv,Wo,bo,g0,be0,g1,be1)
//
// Workspace: 3*(32*2048*512) + 4*(32*64*512) floats ~= 420 MB.
// ============================================================================

typedef __attribute__((ext_vector_type(2))) float v2f;
typedef __attribute__((ext_vector_type(8))) float v8f;

#define B_   32
#define N_   2048
#define DH_  512
#define H_   8
#define HD_  64
#define NI_  64

__device__ __forceinline__ v8f wmma_f32(v2f a, v2f b, v8f c) {
  // D = A(16x4 f32) * B(4x16 f32) + C(16x16 f32)
  return __builtin_amdgcn_wmma_f32_16x16x4_f32(
      /*neg_a=*/false, a, /*neg_b=*/false, b,
      /*c_mod=*/(short)0, c, /*reuse_a=*/false, /*reuse_b=*/false);
}

// gfx1250 async global->LDS copy (16B per lane), tracked by ASYNCcnt.
__device__ __forceinline__ void async_load_b128(unsigned int lds_off,
                                                const void* gaddr) {
  asm volatile("global_load_async_to_lds_b128 %0, %1, off"
               :: "v"(lds_off), "v"(gaddr)
               : "memory");
}
__device__ __forceinline__ void wait_async0() {
  asm volatile("s_wait_asynccnt 0" ::: "memory");
}

// One K-chunk (16) of WMMAs: wave computes a 16(M) x 64(N) strip.
__device__ __forceinline__ void mma_step(const float (*as)[20],
                                         const float (*ws)[20],
                                         int wave, int lm, int kh2,
                                         v8f& c0, v8f& c1, v8f& c2, v8f& c3) {
  #pragma unroll
  for (int kk = 0; kk < 4; ++kk) {
    const int col = kk * 4 + 2 * kh2;
    v2f a, b0, b1, b2, b3;
    a.x  = as[wave * 16 + lm][col]; a.y  = as[wave * 16 + lm][col + 1];
    b0.x = ws[lm][col];             b0.y = ws[lm][col + 1];
    b1.x = ws[16 + lm][col];        b1.y = ws[16 + lm][col + 1];
    b2.x = ws[32 + lm][col];        b2.y = ws[32 + lm][col + 1];
    b3.x = ws[48 + lm][col];        b3.y = ws[48 + lm][col + 1];
    c0 = wmma_f32(a, b0, c0);
    c1 = wmma_f32(a, b1, c1);
    c2 = wmma_f32(a, b2, c2);
    c3 = wmma_f32(a, b3, c3);
  }
}

// ----------------------------------------------------------------------------
// GEMM: C[b,m,n] = epi( sum_k A[b,m,k] * W[n,k] + bias[n] )
// A row-major [s, din] per batch (batch stride strideA; 0 => broadcast),
// W row-major [dout, din]. mode 0: none | 1: relu | 2: Res + relu(.)
// Block = 128 threads (4 waves); WG tile 64(M) x 64(N); wave tile 16 x 64.
// FULL=true : s % 64 == 0 -> unguarded, async double-buffered LDS staging.
// FULL=false: guarded synchronous staging (only used for s==1, PMA seed).
// ----------------------------------------------------------------------------
template <bool FULL>
__global__ __launch_bounds__(128)
void gemm_wmma_k(const float* __restrict__ A, const float* __restrict__ W,
                 const float* __restrict__ bias, const float* __restrict__ Res,
                 float* __restrict__ C, int s, int din, int dout, int strideA,
                 int mode)
{
  // stride 20 floats = 80 B: keeps async B128 stores 16B-aligned and frag
  // reads bank-conflict-free (20*m mod 64 distinct for m = 0..15).
  __shared__ float As[2][64][20];
  __shared__ float Ws[2][64][20];
  const int b    = blockIdx.z;
  const int m0   = blockIdx.x * 64;
  const int n0   = blockIdx.y * 64;
  const int tid  = threadIdx.x;
  const int wave = tid >> 5, lane = tid & 31;
  const int lm   = lane & 15, kh2 = lane >> 4;

  const float* Ab = A + (size_t)b * (size_t)strideA;
  v8f c0 = {}, c1 = {}, c2 = {}, c3 = {};

  if (FULL) {
    // staging map: thread -> (row = tid>>1, two 16B segments)
    const int srow = tid >> 1;
    const int sc0  = (tid & 1) * 8;               // starting column (floats)
    const float* arow = Ab + (size_t)(m0 + srow) * din;
    const float* wrow = W  + (size_t)(n0 + srow) * din;

    // prefetch chunk 0 into buffer 0
    async_load_b128((unsigned int)(size_t)&As[0][srow][sc0],     arow + sc0);
    async_load_b128((unsigned int)(size_t)&As[0][srow][sc0 + 4], arow + sc0 + 4);
    async_load_b128((unsigned int)(size_t)&Ws[0][srow][sc0],     wrow + sc0);
    async_load_b128((unsigned int)(size_t)&Ws[0][srow][sc0 + 4], wrow + sc0 + 4);
    wait_async0();
    __syncthreads();

    int cur = 0;
    for (int k0 = 0; k0 < din; k0 += 16) {
      const int nx = k0 + 16;
      if (nx < din) {                              // prefetch next chunk
        const int nb = cur ^ 1;
        async_load_b128((unsigned int)(size_t)&As[nb][srow][sc0],     arow + nx + sc0);
        async_load_b128((unsigned int)(size_t)&As[nb][srow][sc0 + 4], arow + nx + sc0 + 4);
        async_load_b128((unsigned int)(size_t)&Ws[nb][srow][sc0],     wrow + nx + sc0);
        async_load_b128((unsigned int)(size_t)&Ws[nb][srow][sc0 + 4], wrow + nx + sc0 + 4);
      }
      mma_step(As[cur], Ws[cur], wave, lm, kh2, c0, c1, c2, c3);
      wait_async0();        // my async transfers done (overlapped with WMMAs)
      __syncthreads();      // everyone's done + everyone finished reading cur
      cur ^= 1;
    }
  } else {
    for (int k0 = 0; k0 < din; k0 += 16) {
      for (int li = tid; li < 1024; li += 128) {
        int r = li >> 4, cc = li & 15;
        int gm = m0 + r;
        As[0][r][cc] = (gm < s) ? Ab[(size_t)gm * din + k0 + cc] : 0.f;
        Ws[0][r][cc] = W[(size_t)(n0 + r) * din + k0 + cc];
      }
      __syncthreads();
      mma_step(As[0], Ws[0], wave, lm, kh2, c0, c1, c2, c3);
      __syncthreads();
    }
  }

  // C/D layout: VGPR r -> M = r (lanes 0-15) / r+8 (lanes 16-31), N = lane&15
  #pragma unroll
  for (int r = 0; r < 8; ++r) {
    int m = m0 + wave * 16 + r + kh2 * 8;
    if (!FULL && m >= s) continue;
    size_t rowoff = ((size_t)b * s + m) * (size_t)dout;
    float vals[4] = {c0[r], c1[r], c2[r], c3[r]};
    #pragma unroll
    for (int nt = 0; nt < 4; ++nt) {
      int n = n0 + nt * 16 + lm;
      float acc = vals[nt] + bias[n];
      if (mode >= 1) acc = fmaxf(acc, 0.f);
      if (mode == 2) acc += Res[rowoff + n];
      C[rowoff + n] = acc;
    }
  }
}

// ----------------------------------------------------------------------------
// Fused attention: O[b,m,h*64+c] = q[...] + (softmax(qh khT / sqrt(512)) vh)
// One WG per (16-query tile, head, batch). Full score rows live in dynamic
// LDS (16 x (sk+1) floats: 131 KB for sk=2048, fits the 320 KB WGP LDS).
// Phase 1: QK^T via WMMA. Phase 2: in-LDS row softmax (shuffle reductions).
// Phase 3: A*V via WMMA with a 2-phase cross-wave LDS add.
// ----------------------------------------------------------------------------
__global__ __launch_bounds__(256)
void attn_k(const float* __restrict__ q, const float* __restrict__ k,
            const float* __restrict__ v, float* __restrict__ O,
            int sq, int sk)
{
  extern __shared__ float smem[];
  const int SROW = sk + 1;
  float* S  = smem;                 // 16 x SROW   (scores)
  float* qs = S + 16 * SROW;        // 16 x 68     (q tile, padded)
  float* ot = qs + 16 * 68;         // 16 x 68     (output tile, padded)

  const int b = blockIdx.z, h = blockIdx.y, q0 = blockIdx.x * 16;
  const int tid = threadIdx.x, wave = tid >> 5, lane = tid & 31;
  const int lm = lane & 15, kh2 = lane >> 4;
  const float scale = 0.04419417382415922f;  // 1/sqrt(DH) = 1/sqrt(512)

  // ---- load q tile (zero-pad rows beyond sq) ----
  for (int li = tid; li < 16 * 64; li += 256) {
    int r = li >> 6, cc = li & 63;
    int gm = q0 + r;
    qs[r * 68 + cc] = (gm < sq) ? q[((size_t)(b * sq + gm)) * DH_ + h * HD_ + cc]
                                : 0.f;
  }
  __syncthreads();

  // ---- phase 1: S[m,key] = qh . kh, key tiles round-robin over 8 waves ----
  const int nkt = sk >> 4;
  for (int kt = wave; kt < nkt; kt += 8) {
    v8f c = {};
    const float* kb =
        k + ((size_t)(b * sk + kt * 16 + lm)) * DH_ + h * HD_ + 2 * kh2;
    #pragma unroll
    for (int kc = 0; kc < 16; ++kc) {     // K = 64 = 16 steps of 4
      v2f a, bb;
      a.x = qs[lm * 68 + kc * 4 + 2 * kh2];
      a.y = qs[lm * 68 + kc * 4 + 2 * kh2 + 1];
      bb = *(const v2f*)(kb + kc * 4);    // 8B-aligned pair load
      c = wmma_f32(a, bb, c);
    }
    #pragma unroll
    for (int r = 0; r < 8; ++r)
      S[(r + kh2 * 8) * SROW + kt * 16 + lm] = c[r];
  }
  __syncthreads();

  // ---- phase 2: softmax along keys (16 threads per row, shuffle reduce) ----
  {
    int row = tid >> 4, sub = tid & 15;
    float mx = -1e30f;
    for (int j = sub; j < sk; j += 16) mx = fmaxf(mx, S[row * SROW + j]);
    for (int m2 = 8; m2 >= 1; m2 >>= 1) mx = fmaxf(mx, __shfl_xor(mx, m2, 32));
    float sum = 0.f;
    for (int j = sub; j < sk; j += 16) {
      float e = __expf((S[row * SROW + j] - mx) * scale);
      S[row * SROW + j] = e;
      sum += e;
    }
    for (int m2 = 8; m2 >= 1; m2 >>= 1) sum += __shfl_xor(sum, m2, 32);
    float inv = 1.f / sum;
    for (int j = sub; j < sk; j += 16) S[row * SROW + j] *= inv;
  }
  __syncthreads();

  // ---- phase 3: Ohead(16x64) = S(16xsk) * vh(skx64) ----
  {
    int nt = wave & 3, half = wave >> 2;
    int key0 = half * (sk >> 1);
    int nsteps = sk >> 3;                 // (sk/2) keys / 4 per WMMA
    v8f c = {};
    for (int st = 0; st < nsteps; ++st) {
      int kk = key0 + st * 4;
      v2f a, bb;
      a.x = S[lm * SROW + kk + 2 * kh2];
      a.y = S[lm * SROW + kk + 2 * kh2 + 1];
      const float* vb =
          v + ((size_t)(b * sk + kk + 2 * kh2)) * DH_ + h * HD_ + nt * 16 + lm;
      bb.x = vb[0];
      bb.y = vb[DH_];
      c = wmma_f32(a, bb, c);
    }
    if (half == 0) {
      #pragma unroll
      for (int r = 0; r < 8; ++r) ot[(r + kh2 * 8) * 68 + nt * 16 + lm] = c[r];
    }
    __syncthreads();
    if (half == 1) {
      #pragma unroll
      for (int r = 0; r < 8; ++r) ot[(r + kh2 * 8) * 68 + nt * 16 + lm] += c[r];
    }
    __syncthreads();
  }

  // ---- epilogue: O = q + attn ----
  for (int li = tid; li < 16 * 64; li += 256) {
    int r = li >> 6, cc = li & 63;
    int gm = q0 + r;
    if (gm < sq) {
      size_t idx = ((size_t)(b * sq + gm)) * DH_ + h * HD_ + cc;
      O[idx] = q[idx] + ot[r * 68 + cc];
    }
  }
}

// ----------------------------------------------------------------------------
// LayerNorm over last dim (512). One wave per row; 16 elems/lane.
// ----------------------------------------------------------------------------
__global__ __launch_bounds__(256)
void ln_k(const float* __restrict__ X, const float* __restrict__ g,
          const float* __restrict__ be, float* __restrict__ Y, int rows)
{
  int wave = threadIdx.x >> 5, lane = threadIdx.x & 31;
  int row = blockIdx.x * 8 + wave;
  if (row >= rows) return;
  const float* xr = X + (size_t)row * DH_;
  float xv[16];
  float sum = 0.f;
  #pragma unroll
  for (int i = 0; i < 16; ++i) { xv[i] = xr[i * 32 + lane]; sum += xv[i]; }
  for (int m = 16; m >= 1; m >>= 1) sum += __shfl_xor(sum, m, 32);
  float mean = sum * (1.f / 512.f);
  float s2 = 0.f;
  #pragma unroll
  for (int i = 0; i < 16; ++i) { float d = xv[i] - mean; s2 += d * d; }
  for (int m = 16; m >= 1; m >>= 1) s2 += __shfl_xor(s2, m, 32);
  float inv = rsqrtf(s2 * (1.f / 512.f) + 1e-5f);
  float* yr = Y + (size_t)row * DH_;
  #pragma unroll
  for (int i = 0; i < 16; ++i) {
    int cidx = i * 32 + lane;
    yr[cidx] = (xv[i] - mean) * inv * g[cidx] + be[cidx];
  }
}

// ============================================================================
// Host-side orchestration
// ============================================================================
struct MabW {
  const float *Wq, *bq, *Wk, *bk, *Wv, *bv, *Wo, *bo, *g0, *be0, *g1, *be1;
};

static MabW mab_at(void* const* d_in, int base) {
  MabW w;
  w.Wq = (const float*)d_in[base + 0];  w.bq  = (const float*)d_in[base + 1];
  w.Wk = (const float*)d_in[base + 2];  w.bk  = (const float*)d_in[base + 3];
  w.Wv = (const float*)d_in[base + 4];  w.bv  = (const float*)d_in[base + 5];
  w.Wo = (const float*)d_in[base + 6];  w.bo  = (const float*)d_in[base + 7];
  w.g0 = (const float*)d_in[base + 8];  w.be0 = (const float*)d_in[base + 9];
  w.g1 = (const float*)d_in[base + 10]; w.be1 = (const float*)d_in[base + 11];
  return w;
}

static void launch_gemm(const float* A, const float* W, const float* bias,
                        const float* Res, float* C, int s, int din, int dout,
                        int strideA, int mode, hipStream_t st) {
  dim3 g((s + 63) / 64, dout / 64, B_);
  if ((s & 63) == 0)
    gemm_wmma_k<true><<<g, 128, 0, st>>>(A, W, bias, Res, C, s, din, dout,
                                         strideA, mode);
  else
    gemm_wmma_k<false><<<g, 128, 0, st>>>(A, W, bias, Res, C, s, din, dout,
                                          strideA, mode);
}

// MAB(Q[B,sq,dq], K[B,sk,dk]) -> out[B,sq,512]
static void run_mab(const float* Qsrc, int qBroadcast, int sq, int dq,
                    const float* Ksrc, int sk, int dk, const MabW& w,
                    float* qb, float* kb, float* vb, float* t1, float* t2,
                    float* out, hipStream_t st) {
  launch_gemm(Qsrc, w.Wq, w.bq, nullptr, qb, sq, dq, DH_,
              qBroadcast ? 0 : sq * dq, 0, st);
  launch_gemm(Ksrc, w.Wk, w.bk, nullptr, kb, sk, dk, DH_, sk * dk, 0, st);
  launch_gemm(Ksrc, w.Wv, w.bv, nullptr, vb, sk, dk, DH_, sk * dk, 0, st);
  size_t smem = (size_t)(16 * (sk + 1) + 2 * 16 * 68) * sizeof(float);
  dim3 ga((sq + 15) / 16, H_, B_);
  attn_k<<<ga, 256, smem, st>>>(qb, kb, vb, t1, sq, sk);
  ln_k<<<(B_ * sq + 7) / 8, 256, 0, st>>>(t1, w.g0, w.be0, t1, B_ * sq);
  launch_gemm(t1, w.Wo, w.bo, t1, t2, sq, DH_, DH_, sq * DH_, 2, st);
  ln_k<<<(B_ * sq + 7) / 8, 256, 0, st>>>(t2, w.g1, w.be1, out, B_ * sq);
}

extern "C" void kernel_launch(void* const* d_in, const int* in_sizes, int n_in,
                              void* d_out, int out_size, void* d_ws,
                              size_t ws_size, hipStream_t stream) {
  (void)in_sizes; (void)n_in; (void)out_size; (void)ws_size;

  const float* x  = (const float*)d_in[0];
  const float* Wl = (const float*)d_in[1];
  const float* bl = (const float*)d_in[2];
  const float* I1 = (const float*)d_in[3];
  MabW m10 = mab_at(d_in, 4);
  MabW m11 = mab_at(d_in, 16);
  const float* Ia = (const float*)d_in[28];
  MabW ma0 = mab_at(d_in, 29);
  MabW ma1 = mab_at(d_in, 41);
  const float* Ib = (const float*)d_in[53];
  MabW mb0 = mab_at(d_in, 54);
  MabW mb1 = mab_at(d_in, 66);
  const float* Sp = (const float*)d_in[78];
  MabW mp = mab_at(d_in, 79);

  const size_t BIG = (size_t)B_ * N_ * DH_;    // 33,554,432 floats (134 MB)
  const size_t SM  = (size_t)B_ * NI_ * DH_;   //  1,048,576 floats (4 MB)
  float* W0 = (float*)d_ws;        // current set activations X
  float* W1 = W0 + BIG;            // big scratch
  float* W2 = W1 + BIG;            // big scratch
  float* Hh = W2 + BIG;            // induced features [B,64,512]
  float* SA = Hh + SM;
  float* SB = SA + SM;
  float* SC = SB + SM;
  // total workspace: 3*BIG + 4*SM floats ~= 420 MB

  // Allow >64KB dynamic LDS for the sk=2048 attention variant (~140 KB).
  hipFuncSetAttribute(reinterpret_cast<const void*>(attn_k),
                      hipFuncAttributeMaxDynamicSharedMemorySize, 160 * 1024);

  // h = relu(x Wl^T + bl) : [B, N, 256], stored tightly in W0
  launch_gemm(x, Wl, bl, nullptr, W0, N_, 16, 256, N_ * 16, 1, stream);

  // ISAB1 (X has dim 256)
  run_mab(I1, 1, NI_, DH_, W0, N_, 256, m10, SA, W1, W2, SB, SC, Hh, stream);
  run_mab(W0, 0, N_, 256, Hh, NI_, DH_, m11, W1, SA, SB, W2, W1, W0, stream);
  // ISAB a
  run_mab(Ia, 1, NI_, DH_, W0, N_, DH_, ma0, SA, W1, W2, SB, SC, Hh, stream);
  run_mab(W0, 0, N_, DH_, Hh, NI_, DH_, ma1, W1, SA, SB, W2, W1, W0, stream);
  // ISAB b
  run_mab(Ib, 1, NI_, DH_, W0, N_, DH_, mb0, SA, W1, W2, SB, SC, Hh, stream);
  run_mab(W0, 0, N_, DH_, Hh, NI_, DH_, mb1, W1, SA, SB, W2, W1, W0, stream);
  // PMA: MAB(S[1,1,512] broadcast, X) -> d_out [B,1,512]
  run_mab(Sp, 1, 1, DH_, W0, N_, DH_, mp, SA, W1, W2, SB, SC, (float*)d_out,
          stream);
}